// CompGCNEncoder_52467320487976
// MI455X (gfx1250) — compile-verified
//
#include <hip/hip_runtime.h>
#include <hip/hip_bf16.h>

#define DD 128

typedef float v2f __attribute__((ext_vector_type(2)));
typedef float v8f __attribute__((ext_vector_type(8)));
typedef int   v4i __attribute__((ext_vector_type(4)));

// ---- CDNA5 async-to-LDS path (ASYNCcnt), guarded so compile stays green ----
// Probe round 2 revealed the builtin's prototype: (v4i AS1*, v4i AS3*, imm off, imm cpol).
#if defined(__HIP_DEVICE_COMPILE__) && defined(__gfx1250__) && \
    __has_builtin(__builtin_amdgcn_global_load_async_to_lds_b128)
#define HAVE_ASYNC_LDS 1
// INST_OFFSET is added to BOTH the global and the LDS address (ISA 08 §4.4).
#define ASYNC_B128(g, l, off)                                              \
    __builtin_amdgcn_global_load_async_to_lds_b128(                        \
        (__attribute__((address_space(1))) v4i*)(g),                       \
        (__attribute__((address_space(3))) v4i*)(l), (off), 0)
#if __has_builtin(__builtin_amdgcn_s_wait_asynccnt)
#define WAIT_ASYNC(n) __builtin_amdgcn_s_wait_asynccnt(n)
#else
#define WAIT_ASYNC(n) asm volatile("s_wait_asynccnt " #n ::: "memory")
#endif
#endif

// ---------------------------------------------------------------- degrees
__global__ void deg_kernel(const int* __restrict__ ei0, float* degA, float* degB, int E) {
    int e = blockIdx.x * 256 + threadIdx.x;
    if (e < E) {
        atomicAdd(degA + ei0[e], 1.0f);       // in-direction:  hist over src half
        atomicAdd(degB + ei0[E + e], 1.0f);   // out-direction: hist over dst half
    }
}

__global__ void dinv_kernel(float* degA, float* degB, int n) {
    int i = blockIdx.x * 256 + threadIdx.x;
    if (i < n) {
        float a = degA[i]; degA[i] = (a > 0.0f) ? rsqrtf(a) : 0.0f;
        float b = degB[i]; degB[i] = (b > 0.0f) ? rsqrtf(b) : 0.0f;
    }
}

__global__ void norm_kernel(const int* __restrict__ ei0, const int* __restrict__ ei1,
                            const float* __restrict__ degA, const float* __restrict__ degB,
                            float* normv, int E, int twoE) {
    int e = blockIdx.x * 256 + threadIdx.x;
    if (e < twoE) {
        const float* dv = (e < E) ? degA : degB;
        normv[e] = dv[ei0[e]] * dv[ei1[e]];
    }
}

// ---------------------------------------------------------------- sort by type
__global__ void hist_kernel(const int* __restrict__ et, int* cnt, int n) {
    int e = blockIdx.x * 256 + threadIdx.x;
    if (e < n) atomicAdd(cnt + et[e], 1);
}

__global__ void scan_kernel(const int* __restrict__ cnt, int* off) {
    if (threadIdx.x == 0) {
        int s = 0;
        for (int t = 0; t < 200; ++t) { off[t] = s; s += cnt[t]; }
        off[200] = s;
    }
}

__global__ void scatter_kernel(const int* __restrict__ et, const int* __restrict__ off,
                               int* cursor, int* sorted, int n) {
    int e = blockIdx.x * 256 + threadIdx.x;
    if (e < n) {
        int t = et[e];
        int p = off[t] + atomicAdd(cursor + t, 1);
        sorted[p] = e;
    }
}

// ---------------------------------------------------------------- fused M_t = Circ(r_t) @ W
// M_t[j][c] = sum_k r_t[(j+k)%128] * W[k][c]   (folds ccorr into the weight matrix)
__global__ void build_M(const float* __restrict__ rel_embed, const float* __restrict__ loop_rel,
                        const float* __restrict__ w_in, const float* __restrict__ w_out,
                        const float* __restrict__ w_loop, float* __restrict__ Mm) {
    __shared__ float r[DD];
    int t = blockIdx.x;  // 0..200
    const float* rsrc = (t < 200) ? (rel_embed + (size_t)t * DD) : loop_rel;
    const float* W    = (t < 100) ? w_in : ((t < 200) ? w_out : w_loop);
    if (threadIdx.x < DD) r[threadIdx.x] = rsrc[threadIdx.x];
    __syncthreads();
    int c  = threadIdx.x & 127;
    int j0 = threadIdx.x >> 7;     // 0 or 1
    float* Mt = Mm + (size_t)t * DD * DD;
    for (int j = j0; j < DD; j += 2) {
        float s = 0.0f;
        for (int k = 0; k < DD; ++k) s += r[(j + k) & 127] * W[k * DD + c];
        Mt[j * DD + c] = s;
    }
}

// ---------------------------------------------------------------- WMMA tile helper
// One wave computes a 16x16 output tile: C = A(16x128) * B(128x16) via 32x V_WMMA_F32_16X16X4_F32
// A rows come from LDS Xt (stride 132 to dodge bank conflicts), B from LDS-resident M.
__device__ __forceinline__ v8f wmma_row_tile(const float* Xt, const float* Mlds,
                                             int lane, int wave) {
    v8f acc = {0.f, 0.f, 0.f, 0.f, 0.f, 0.f, 0.f, 0.f};
    const int krow = 2 * (lane >> 4);             // 0 (lanes 0-15) or 2 (lanes 16-31)
    const int mrow = lane & 15;
    const int ncol = (wave << 4) + (lane & 15);
    for (int k0 = 0; k0 < DD; k0 += 4) {
        v2f a, b;
        a.x = Xt[mrow * 132 + k0 + krow];
        a.y = Xt[mrow * 132 + k0 + krow + 1];
        b.x = Mlds[(k0 + krow) * DD + ncol];
        b.y = Mlds[(k0 + krow + 1) * DD + ncol];
        acc = __builtin_amdgcn_wmma_f32_16x16x4_f32(false, a, false, b,
                                                    (short)0, acc, false, false);
    }
    return acc;
}

// ---------------------------------------------------------------- self-loop: accum = x @ M_loop
__global__ void loop_gemm(const float* __restrict__ x, const float* __restrict__ Mm,
                          float* __restrict__ accum, int n_ent) {
    extern __shared__ float Mlds[];               // 128*128 f32 = 64 KB
    __shared__ float Xt[16 * 132];
    const float* Msrc = Mm + (size_t)200 * DD * DD;
    for (int i = threadIdx.x; i < DD * DD / 4; i += 256)
        ((float4*)Mlds)[i] = ((const float4*)Msrc)[i];

    const int r0 = blockIdx.x * 16;
    {   // 16 threads per row, 8 floats each (coalesced 512B rows)
        int m = threadIdx.x >> 4, p = threadIdx.x & 15;
        float4 v0 = {0,0,0,0}, v1 = {0,0,0,0};
        if (r0 + m < n_ent) {
            const float4* s4 = (const float4*)(x + (size_t)(r0 + m) * DD) + p * 2;
            v0 = s4[0]; v1 = s4[1];
        }
        ((float4*)(Xt + m * 132 + p * 8))[0] = v0;
        ((float4*)(Xt + m * 132 + p * 8))[1] = v1;
    }
    __syncthreads();

    const int lane = threadIdx.x & 31, wave = threadIdx.x >> 5;
    v8f acc = wmma_row_tile(Xt, Mlds, lane, wave);
    const int ncol = (wave << 4) + (lane & 15);
    const int half = lane >> 4;
    for (int v = 0; v < 8; ++v) {
        int m = v + (half << 3);
        if (r0 + m < n_ent) accum[(size_t)(r0 + m) * DD + ncol] = acc[v];
    }
}

// ---------------------------------------------------------------- edge message kernel
// blockIdx.x = relation type (M_t pinned in 64KB LDS), blockIdx.y strides over 16-edge chunks.
// Async variant: X tiles are double-buffered and staged with GLOBAL_LOAD_ASYNC_TO_LDS_B128;
// the M_t copy is also async, drained by the first s_wait_asynccnt.
__global__ void edge_kernel(const float* __restrict__ x,
                            const int* __restrict__ ei0, const int* __restrict__ ei1,
                            const float* __restrict__ normv,
                            const int* __restrict__ sorted,
                            const int* __restrict__ off, const int* __restrict__ cnt,
                            const float* __restrict__ Mm, float* __restrict__ accum) {
    extern __shared__ float Mlds[];               // 64 KB
    __shared__ float Xt[2][16 * 132];             // ping-pong tiles
    __shared__ float nrm_s[2][16];
    __shared__ int   row_s[2][16];

    const int t    = blockIdx.x;
    const int lane = threadIdx.x & 31;
    const int wave = threadIdx.x >> 5;
    const int base = off[t];
    const int n    = cnt[t];
    const int nchunk = (n + 15) >> 4;

#ifdef HAVE_ASYNC_LDS
    {   // async M_t copy: 16 B128 per thread, lands before first s_wait_asynccnt
        const float4* Msrc4 = (const float4*)(Mm + (size_t)t * DD * DD);
        float4* Ml4 = (float4*)Mlds;
        for (int i = threadIdx.x; i < DD * DD / 4; i += 256)
            ASYNC_B128(Msrc4 + i, Ml4 + i, 0);
    }
    const int m = threadIdx.x >> 4, p = threadIdx.x & 15;

    // stage chunk c0 into buffer 0
    int c0 = blockIdx.y;
    if (c0 < nchunk) {
        int idx = (c0 << 4) + m, src = 0;
        if (idx < n) {
            int e = sorted[base + idx];
            src = ei1[e];
            if (p == 0) { row_s[0][m] = ei0[e]; nrm_s[0][m] = normv[e]; }
        } else if (p == 0) { row_s[0][m] = 0; nrm_s[0][m] = 0.0f; }
        ASYNC_B128(x + (size_t)src * DD + p * 8, &Xt[0][m * 132 + p * 8], 0);
        ASYNC_B128(x + (size_t)src * DD + p * 8, &Xt[0][m * 132 + p * 8], 16);
    }

    int buf = 0;
    for (int c = c0; c < nchunk; c += gridDim.y, buf ^= 1) {
        const int nc = c + gridDim.y;          // uniform across the block
        if (nc < nchunk) {                     // stage next chunk into buf^1
            int idx = (nc << 4) + m, src = 0;
            if (idx < n) {
                int e = sorted[base + idx];
                src = ei1[e];
                if (p == 0) { row_s[buf ^ 1][m] = ei0[e]; nrm_s[buf ^ 1][m] = normv[e]; }
            } else if (p == 0) { row_s[buf ^ 1][m] = 0; nrm_s[buf ^ 1][m] = 0.0f; }
            ASYNC_B128(x + (size_t)src * DD + p * 8, &Xt[buf ^ 1][m * 132 + p * 8], 0);
            ASYNC_B128(x + (size_t)src * DD + p * 8, &Xt[buf ^ 1][m * 132 + p * 8], 16);
            WAIT_ASYNC(2);                     // drain everything except the 2 just issued
        } else {
            WAIT_ASYNC(0);
        }
        __syncthreads();                       // all waves' async data visible

        v8f acc = wmma_row_tile(Xt[buf], Mlds, lane, wave);

        const int ncol = (wave << 4) + (lane & 15);
        const int half = lane >> 4;
        for (int v = 0; v < 8; ++v) {
            int mm = v + (half << 3);
            float nm = nrm_s[buf][mm];
            if (nm != 0.0f)
                atomicAdd(accum + (size_t)row_s[buf][mm] * DD + ncol, acc[v] * nm);
        }
        __syncthreads();                       // buf may be re-staged next iteration
    }
#else
    // -------- synchronous fallback (round-1 path) --------
    for (int i = threadIdx.x; i < DD * DD / 4; i += 256)
        ((float4*)Mlds)[i] = ((const float4*)(Mm + (size_t)t * DD * DD))[i];
    __syncthreads();

    for (int c = blockIdx.y; c < nchunk; c += gridDim.y) {
        if (threadIdx.x < 16) {
            int idx = (c << 4) + threadIdx.x;
            if (idx < n) {
                int e = sorted[base + idx];
                row_s[0][threadIdx.x] = ei0[e];
                row_s[1][threadIdx.x] = ei1[e];       // reuse as src slot
                nrm_s[0][threadIdx.x] = normv[e];
            } else {
                row_s[0][threadIdx.x] = 0; row_s[1][threadIdx.x] = -1;
                nrm_s[0][threadIdx.x] = 0.0f;
            }
        }
        __syncthreads();
        {
            int m = threadIdx.x >> 4, p = threadIdx.x & 15;
            int sr = row_s[1][m];
            float4 v0 = {0,0,0,0}, v1 = {0,0,0,0};
            if (sr >= 0) {
                const float4* s4 = (const float4*)(x + (size_t)sr * DD) + p * 2;
                v0 = s4[0]; v1 = s4[1];
            }
            ((float4*)(&Xt[0][m * 132 + p * 8]))[0] = v0;
            ((float4*)(&Xt[0][m * 132 + p * 8]))[1] = v1;
        }
        __syncthreads();

        v8f acc = wmma_row_tile(Xt[0], Mlds, lane, wave);

        const int ncol = (wave << 4) + (lane & 15);
        const int half = lane >> 4;
        for (int v = 0; v < 8; ++v) {
            int mm = v + (half << 3);
            float nm = nrm_s[0][mm];
            if (nm != 0.0f)
                atomicAdd(accum + (size_t)row_s[0][mm] * DD + ncol, acc[v] * nm);
        }
        __syncthreads();
    }
#endif
}

// ---------------------------------------------------------------- batch norm
__global__ void bn_stats(const float* __restrict__ accum, const float* __restrict__ bias,
                         float* gsum, float* gsq, int total) {
    // stride is a multiple of 128 -> each thread stays on column (threadIdx.x & 127)
    const int c = threadIdx.x & 127;
    const float b = bias[c];
    float s = 0.0f, q = 0.0f;
    for (size_t i = (size_t)blockIdx.x * 256 + threadIdx.x; i < (size_t)total;
         i += (size_t)gridDim.x * 256) {
        float y = accum[i] * (1.0f / 3.0f) + b;
        s += y; q += y * y;
    }
    __shared__ float ss[256], qq[256];
    ss[threadIdx.x] = s; qq[threadIdx.x] = q;
    __syncthreads();
    if (threadIdx.x < 128) {
        atomicAdd(gsum + c, ss[threadIdx.x] + ss[threadIdx.x + 128]);
        atomicAdd(gsq  + c, qq[threadIdx.x] + qq[threadIdx.x + 128]);
    }
}

__global__ void bn_finalize(const float* gsum, const float* gsq,
                            float* meanv, float* rstdv, int n_rows) {
    int c = threadIdx.x;
    if (c < 128) {
        float inv = 1.0f / (float)n_rows;
        float mu  = gsum[c] * inv;
        float var = gsq[c] * inv - mu * mu;
        meanv[c] = mu;
        rstdv[c] = rsqrtf(var + 1e-5f);
    }
}

__global__ void bn_apply(const float* __restrict__ accum, const float* __restrict__ bias,
                         const float* __restrict__ gamma, const float* __restrict__ beta,
                         const float* __restrict__ meanv, const float* __restrict__ rstdv,
                         float* __restrict__ out, int total) {
    int i = blockIdx.x * 256 + threadIdx.x;
    if (i < total) {
        int c = i & 127;
        float y = accum[i] * (1.0f / 3.0f) + bias[c];
        out[i] = gamma[c] * (y - meanv[c]) * rstdv[c] + beta[c];
    }
}

// ---------------------------------------------------------------- rel_out = rel_embed @ w_rel
__global__ void rel_out_kernel(const float* __restrict__ rel_embed,
                               const float* __restrict__ w_rel,
                               float* __restrict__ out, int n_rel) {
    int i = blockIdx.x * 256 + threadIdx.x;
    if (i < n_rel * DD) {
        int r = i >> 7, c = i & 127;
        float s = 0.0f;
        for (int k = 0; k < DD; ++k) s += rel_embed[r * DD + k] * w_rel[k * DD + c];
        out[i] = s;
    }
}

// ---------------------------------------------------------------- host
extern "C" void kernel_launch(void* const* d_in, const int* in_sizes, int n_in,
                              void* d_out, int out_size, void* d_ws, size_t ws_size,
                              hipStream_t stream) {
    const float* x        = (const float*)d_in[0];
    const int*   ei       = (const int*)d_in[1];
    const int*   et       = (const int*)d_in[2];
    const float* rel      = (const float*)d_in[3];
    const float* w_in     = (const float*)d_in[4];
    const float* w_out    = (const float*)d_in[5];
    const float* w_loop   = (const float*)d_in[6];
    const float* w_rel    = (const float*)d_in[7];
    const float* loop_rel = (const float*)d_in[8];
    const float* bias     = (const float*)d_in[9];
    const float* gamma    = (const float*)d_in[10];
    const float* beta     = (const float*)d_in[11];

    const int twoE  = in_sizes[2];          // 2,000,000
    const int E     = twoE / 2;
    const int n_ent = in_sizes[0] / DD;     // 50,000
    const int* ei0  = ei;                   // edge_index row 0 (targets)
    const int* ei1  = ei + twoE;            // edge_index row 1 (sources)

    char* w = (char*)d_ws;
    size_t cur = 0;
    auto alloc = [&](size_t bytes) -> void* {
        void* p = (void*)(w + cur);
        cur = (cur + bytes + 255) & ~(size_t)255;
        return p;
    };
    float* degA   = (float*)alloc((size_t)n_ent * 4);
    float* degB   = (float*)alloc((size_t)n_ent * 4);
    int*   cnt    = (int*)alloc(200 * 4);
    int*   cursor = (int*)alloc(200 * 4);
    int*   offp   = (int*)alloc(201 * 4);
    float* gsum   = (float*)alloc(128 * 4);
    float* gsq    = (float*)alloc(128 * 4);
    float* meanv  = (float*)alloc(128 * 4);
    float* rstdv  = (float*)alloc(128 * 4);
    size_t zero_bytes = cur;                                   // zero everything above
    float* normv  = (float*)alloc((size_t)twoE * 4);
    int*   sorted = (int*)alloc((size_t)twoE * 4);
    float* Mm     = (float*)alloc((size_t)201 * DD * DD * 4);  // 13.2 MB
    float* accum  = (float*)alloc((size_t)n_ent * DD * 4);     // 25.6 MB

    (void)hipMemsetAsync(d_ws, 0, zero_bytes, stream);

    deg_kernel   <<<(E + 255) / 256, 256, 0, stream>>>(ei0, degA, degB, E);
    dinv_kernel  <<<(n_ent + 255) / 256, 256, 0, stream>>>(degA, degB, n_ent);
    norm_kernel  <<<(twoE + 255) / 256, 256, 0, stream>>>(ei0, ei1, degA, degB, normv, E, twoE);

    hist_kernel   <<<(twoE + 255) / 256, 256, 0, stream>>>(et, cnt, twoE);
    scan_kernel   <<<1, 32, 0, stream>>>(cnt, offp);
    scatter_kernel<<<(twoE + 255) / 256, 256, 0, stream>>>(et, offp, cursor, sorted, twoE);

    build_M<<<201, 256, 0, stream>>>(rel, loop_rel, w_in, w_out, w_loop, Mm);

    loop_gemm<<<(n_ent + 15) / 16, 256, DD * DD * 4, stream>>>(x, Mm, accum, n_ent);

    dim3 eg(200, 16);
    edge_kernel<<<eg, 256, DD * DD * 4, stream>>>(x, ei0, ei1, normv, sorted, offp, cnt, Mm, accum);

    const int total = n_ent * DD;
    bn_stats   <<<1024, 256, 0, stream>>>(accum, bias, gsum, gsq, total);
    bn_finalize<<<1, 128, 0, stream>>>(gsum, gsq, meanv, rstdv, n_ent);
    bn_apply   <<<(total + 255) / 256, 256, 0, stream>>>(accum, bias, gamma, beta, meanv, rstdv,
                                                         (float*)d_out, total);
    rel_out_kernel<<<(200 * DD + 255) / 256, 256, 0, stream>>>(rel, w_rel,
                                                               (float*)d_out + (size_t)total, 200);
}